// Attn_decoder_rnn_30288109371581
// MI455X (gfx1250) — compile-verified
//
#include <hip/hip_runtime.h>
#include <math.h>

typedef float v2f __attribute__((ext_vector_type(2)));
typedef float v4f __attribute__((ext_vector_type(4)));
typedef float v8f __attribute__((ext_vector_type(8)));

#define Bx 64
#define Hx 1024
#define Sx 2048
#define VOCAB 50257
#define NSPLIT 16
#define SPS (Sx / NSPLIT)   // 128 s-values per split
#define LDSW 36             // padded LDS row stride (floats)

// ---------------------------------------------------------------------------
// Embedding gather: xs[b][k] = emb_W[seq[b]][k]
// ---------------------------------------------------------------------------
__global__ __launch_bounds__(256) void gather_x(const int* __restrict__ seq,
                                                const float* __restrict__ embW,
                                                float* __restrict__ xs) {
  int b = blockIdx.x;
  size_t row = (size_t)seq[b] * Hx;
  for (int k = threadIdx.x; k < Hx; k += 256)
    xs[b * Hx + k] = embW[row + k];
}

// ---------------------------------------------------------------------------
// Dual-source GEMM: O[m,n] = sum_k X0[m,k]*W0[n,k] (k<K0)
//                          + sum_k X1[m,k]*W1[n,k] (k<K1)  + bias0[n]+bias1[n]
// M fixed = 64 (batch). Each 256-thread block computes a 64x64 tile of O
// using V_WMMA_F32_16X16X4_F32. act: 0=none, 1=tanh.
// ---------------------------------------------------------------------------
__global__ __launch_bounds__(256) void gemm_dual(
    const float* __restrict__ X0, const float* __restrict__ W0, int ldw0, int K0,
    const float* __restrict__ X1, const float* __restrict__ W1, int ldw1, int K1,
    const float* __restrict__ bias0, const float* __restrict__ bias1,
    float* __restrict__ O, long ldo, int N, int act) {
  __shared__ float Xs[64 * LDSW];
  __shared__ float Ws[64 * LDSW];

  int t = threadIdx.x;
  int wave = t >> 5, lane = t & 31;
  int lh = lane & 15;
  int ko = (lane >> 4) << 1;            // 0 or 2 (K split across lane halves)
  int mt = wave & 3;                    // M sub-tile 0..3
  int nt0 = (wave >> 2) << 1;           // N sub-tile base: 0 or 2
  int n0 = blockIdx.x * 64;

  v8f acc0 = {0.f, 0.f, 0.f, 0.f, 0.f, 0.f, 0.f, 0.f};
  v8f acc1 = {0.f, 0.f, 0.f, 0.f, 0.f, 0.f, 0.f, 0.f};

  int Ktot = K0 + K1;
  for (int kc = 0; kc < Ktot; kc += 32) {
    const float* Xsrc; const float* Wsrc; int ldw, ldx, kloc;
    if (kc < K0) { Xsrc = X0; Wsrc = W0; ldw = ldw0; ldx = K0; kloc = kc; }
    else         { Xsrc = X1; Wsrc = W1; ldw = ldw1; ldx = K1; kloc = kc - K0; }

    // Cooperative load of 64x32 X and W tiles into padded LDS.
#pragma unroll
    for (int i = 0; i < 2; ++i) {
      int f   = t + i * 256;            // 0..511 float4 slots
      int row = f >> 3;                 // 0..63
      int c4  = (f & 7) << 2;           // 0,4,...,28
      v4f xv = *(const v4f*)(Xsrc + (size_t)row * ldx + kloc + c4);
      *(v4f*)(&Xs[row * LDSW + c4]) = xv;
      int n = n0 + row;
      v4f wv = {0.f, 0.f, 0.f, 0.f};
      if (n < N)
        wv = __builtin_nontemporal_load((const v4f*)(Wsrc + (size_t)n * ldw + kloc + c4));
      *(v4f*)(&Ws[row * LDSW + c4]) = wv;
    }
    __syncthreads();

    const float* xrow  = &Xs[(mt * 16 + lh) * LDSW];
    const float* wrow0 = &Ws[(nt0 * 16 + lh) * LDSW];
    const float* wrow1 = &Ws[((nt0 + 1) * 16 + lh) * LDSW];
#pragma unroll
    for (int kk = 0; kk < 32; kk += 4) {
      v2f a;  a.x  = xrow[kk + ko];  a.y  = xrow[kk + ko + 1];
      v2f b0; b0.x = wrow0[kk + ko]; b0.y = wrow0[kk + ko + 1];
      acc0 = __builtin_amdgcn_wmma_f32_16x16x4_f32(false, a, false, b0,
                                                   (short)0, acc0, false, false);
      v2f b1; b1.x = wrow1[kk + ko]; b1.y = wrow1[kk + ko + 1];
      acc1 = __builtin_amdgcn_wmma_f32_16x16x4_f32(false, a, false, b1,
                                                   (short)0, acc1, false, false);
    }
    __syncthreads();
  }

  // Epilogue: C/D layout — VGPR r: lanes 0-15 -> M=r, lanes 16-31 -> M=r+8.
  int nA = n0 + nt0 * 16 + lh;
  int nB = nA + 16;
  float bA = 0.f, bB = 0.f;
  if (nA < N) { if (bias0) bA += bias0[nA]; if (bias1) bA += bias1[nA]; }
  if (nB < N) { if (bias0) bB += bias0[nB]; if (bias1) bB += bias1[nB]; }
  int mrow = mt * 16 + ((lane >> 4) << 3);
#pragma unroll
  for (int r = 0; r < 8; ++r) {
    int m = mrow + r;
    if (nA < N) {
      float v = acc0[r] + bA; if (act == 1) v = tanhf(v);
      O[(size_t)m * ldo + nA] = v;
    }
    if (nB < N) {
      float v = acc1[r] + bB; if (act == 1) v = tanhf(v);
      O[(size_t)m * ldo + nB] = v;
    }
  }
}

// ---------------------------------------------------------------------------
// LSTM pointwise: gates [64,4096] (i,f,g,o) + c0 -> h, c
// ---------------------------------------------------------------------------
__device__ __forceinline__ float sigm(float x) { return 1.f / (1.f + expf(-x)); }

__global__ __launch_bounds__(256) void lstm_pointwise(
    const float* __restrict__ gates, const float* __restrict__ c0,
    float* __restrict__ h_out, float* __restrict__ c_out, float* __restrict__ h_s) {
  int idx = blockIdx.x * 256 + threadIdx.x;       // 0..65535
  int b = idx >> 10, k = idx & 1023;
  const float* g = gates + (size_t)b * 4096;
  float gi = sigm(g[k]);
  float gf = sigm(g[Hx + k]);
  float gg = tanhf(g[2 * Hx + k]);
  float go = sigm(g[3 * Hx + k]);
  float c = gf * c0[idx] + gi * gg;
  float h = go * tanhf(c);
  c_out[idx] = c;
  h_out[idx] = h;
  h_s[idx]   = h;
}

// ---------------------------------------------------------------------------
// Flash-style attention partial: one block per (b, split of S).
// Single pass over encoder_outputs: raw energies to scratch, online-softmax
// rescaled context accumulator + running (m, l) per split.
// ---------------------------------------------------------------------------
__global__ __launch_bounds__(256) void attn_partial(
    const float* __restrict__ h, const float* __restrict__ enc,
    float* __restrict__ energies, float* __restrict__ m_part,
    float* __restrict__ l_part, float* __restrict__ ctx_part) {
  int b = blockIdx.x, split = blockIdx.y;
  __shared__ float hs[Hx];
  __shared__ float encT[8][Hx];
  __shared__ float eT[8];

  int t = threadIdx.x;
  for (int k = t; k < Hx; k += 256) hs[k] = h[b * Hx + k];
  __syncthreads();

  int wave = t >> 5, lane = t & 31;
  float m = -1e30f, l = 0.f;
  float a0 = 0.f, a1 = 0.f, a2 = 0.f, a3 = 0.f;
  int sBeg = split * SPS;

  for (int s0 = sBeg; s0 < sBeg + SPS; s0 += 8) {
    int s = s0 + wave;
    const float* er = enc + ((size_t)s * Bx + b) * Hx;
    float dot = 0.f;
#pragma unroll
    for (int j = 0; j < 8; ++j) {
      int k = (lane << 2) + (j << 7);
      v4f v = __builtin_nontemporal_load((const v4f*)(er + k));
      *(v4f*)(&encT[wave][k]) = v;
      dot += hs[k] * v.x + hs[k + 1] * v.y + hs[k + 2] * v.z + hs[k + 3] * v.w;
    }
#pragma unroll
    for (int off = 16; off > 0; off >>= 1) dot += __shfl_xor(dot, off, 32);
    if (lane == 0) { eT[wave] = dot; energies[b * Sx + s] = dot; }
    __syncthreads();

    float e[8]; float mn = m;
#pragma unroll
    for (int w = 0; w < 8; ++w) { e[w] = eT[w]; mn = fmaxf(mn, e[w]); }
    float scale = expf(m - mn);
    a0 *= scale; a1 *= scale; a2 *= scale; a3 *= scale; l *= scale;
#pragma unroll
    for (int w = 0; w < 8; ++w) {
      float p = expf(e[w] - mn);
      l += p;
      a0 += p * encT[w][t];
      a1 += p * encT[w][t + 256];
      a2 += p * encT[w][t + 512];
      a3 += p * encT[w][t + 768];
    }
    m = mn;
    __syncthreads();
  }

  float* cp = ctx_part + ((size_t)b * NSPLIT + split) * Hx;
  cp[t] = a0; cp[t + 256] = a1; cp[t + 512] = a2; cp[t + 768] = a3;
  if (t == 0) { m_part[b * NSPLIT + split] = m; l_part[b * NSPLIT + split] = l; }
}

// ---------------------------------------------------------------------------
// Combine splits: global max/sum, normalized context + attn weights out.
// ---------------------------------------------------------------------------
__global__ __launch_bounds__(256) void attn_combine(
    const float* __restrict__ m_part, const float* __restrict__ l_part,
    const float* __restrict__ ctx_part, const float* __restrict__ energies,
    float* __restrict__ ctx, float* __restrict__ attn_out) {
  int b = blockIdx.x, t = threadIdx.x;
  __shared__ float wgt[NSPLIT];
  __shared__ float Msh, Lsh;
  if (t == 0) {
    float M = -1e30f;
    for (int i = 0; i < NSPLIT; ++i) M = fmaxf(M, m_part[b * NSPLIT + i]);
    float L = 0.f;
    for (int i = 0; i < NSPLIT; ++i) {
      float w = expf(m_part[b * NSPLIT + i] - M);
      wgt[i] = w;
      L += l_part[b * NSPLIT + i] * w;
    }
    Msh = M; Lsh = L;
  }
  __syncthreads();
  float invL = 1.f / Lsh, M = Msh;
  for (int hh = t; hh < Hx; hh += 256) {
    float s = 0.f;
    for (int i = 0; i < NSPLIT; ++i)
      s += ctx_part[((size_t)b * NSPLIT + i) * Hx + hh] * wgt[i];
    ctx[b * Hx + hh] = s * invL;
  }
  for (int s = t; s < Sx; s += 256)
    attn_out[b * Sx + s] = expf(energies[b * Sx + s] - M) * invL;
}

// ---------------------------------------------------------------------------
// Output layout in d_out (floats): output | h | c | attn
// ---------------------------------------------------------------------------
#define OUT_OFF  0
#define H_OFF    (Bx * VOCAB)                 // 3216448
#define C_OFF    (H_OFF + Bx * Hx)            // 3281984
#define ATTN_OFF (C_OFF + Bx * Hx)            // 3347520

extern "C" void kernel_launch(void* const* d_in, const int* in_sizes, int n_in,
                              void* d_out, int out_size, void* d_ws, size_t ws_size,
                              hipStream_t stream) {
  const int*   seq  = (const int*)d_in[0];
  const float* h0   = (const float*)d_in[1];
  const float* c0   = (const float*)d_in[2];
  const float* enc  = (const float*)d_in[3];
  const float* embW = (const float*)d_in[4];
  const float* W_ih = (const float*)d_in[5];
  const float* W_hh = (const float*)d_in[6];
  const float* b_ih = (const float*)d_in[7];
  const float* b_hh = (const float*)d_in[8];
  const float* cW   = (const float*)d_in[9];
  const float* cb   = (const float*)d_in[10];
  const float* outW = (const float*)d_in[11];
  const float* outb = (const float*)d_in[12];
  float* out = (float*)d_out;

  float* ws     = (float*)d_ws;
  float* xs     = ws;                    // 65536
  float* gates  = xs + 65536;            // 262144
  float* h_s    = gates + 262144;        // 65536
  float* ctx_s  = h_s + 65536;           // 65536
  float* energ  = ctx_s + 65536;         // 131072
  float* m_part = energ + 131072;        // 1024
  float* l_part = m_part + 1024;         // 1024
  float* ctx_p  = l_part + 1024;         // 1048576
  float* co_s   = ctx_p + 1048576;       // 65536

  gather_x<<<Bx, 256, 0, stream>>>(seq, embW, xs);

  // gates = x @ W_ih^T + h0 @ W_hh^T + b_ih + b_hh    [64, 4096]
  gemm_dual<<<4096 / 64, 256, 0, stream>>>(xs, W_ih, Hx, Hx,
                                           h0, W_hh, Hx, Hx,
                                           b_ih, b_hh, gates, 4096, 4096, 0);

  lstm_pointwise<<<(Bx * Hx) / 256, 256, 0, stream>>>(gates, c0,
                                                      out + H_OFF, out + C_OFF, h_s);

  attn_partial<<<dim3(Bx, NSPLIT), 256, 0, stream>>>(h_s, enc, energ,
                                                     m_part, l_part, ctx_p);
  attn_combine<<<Bx, 256, 0, stream>>>(m_part, l_part, ctx_p, energ,
                                       ctx_s, out + ATTN_OFF);

  // concat_out = tanh([h | ctx] @ concat_W^T + concat_b)   [64, 1024]
  gemm_dual<<<1024 / 64, 256, 0, stream>>>(h_s, cW, 2 * Hx, Hx,
                                           ctx_s, cW + Hx, 2 * Hx, Hx,
                                           cb, nullptr, co_s, 1024, 1024, 1);

  // output = concat_out @ out_W^T + out_b   [64, 50257]
  gemm_dual<<<(VOCAB + 63) / 64, 256, 0, stream>>>(co_s, outW, Hx, Hx,
                                                   nullptr, nullptr, 0, 0,
                                                   outb, nullptr, out + OUT_OFF,
                                                   VOCAB, VOCAB, 0);
}